// TorchFEMMesh_15556371546735
// MI455X (gfx1250) — compile-verified
//
#include <hip/hip_runtime.h>
#include <math.h>

// ---------------- problem constants (M=N=80 mesh) ----------------
#define NPAD   6400      // padded free-dof count (6399 -> 6400)
#define BB     64        // Cholesky block size
#define LDP    68        // LDS tile stride: 16B-aligned rows, stride%64==4 -> conflict-free
#define NBLK   (NPAD/BB) // 100

typedef __attribute__((ext_vector_type(2))) float v2f;
typedef __attribute__((ext_vector_type(8))) float v8f;

// ---------------- small setup kernels ----------------
__global__ void fill_i32(int* p, int val, int n) {
  int i = blockIdx.x * blockDim.x + threadIdx.x;
  if (i < n) p[i] = val;
}

__global__ void scatter_map(int* mp, const int* __restrict__ idx, int n) {
  int i = blockIdx.x * blockDim.x + threadIdx.x;
  if (i < n) mp[idx[i]] = i;
}

__global__ void sigma_kernel(const float* __restrict__ mask, float* __restrict__ sigma, int E) {
  int e = blockIdx.x * blockDim.x + threadIdx.x;
  if (e < E) sigma[e] = 0.001f + 0.999f * mask[e >> 1];
}

__global__ void zero_K(float* __restrict__ K, float* __restrict__ b, int nfree) {
  int i = blockIdx.x * blockDim.x + threadIdx.x;
  if (i < NPAD * NPAD) {
    int r = i / NPAD, c = i - r * NPAD;
    K[i] = (r == c && r >= nfree) ? 1.0f : 0.0f;  // identity on padded dofs
  }
  if (i < NPAD) b[i] = 0.0f;
}

// scatter-assemble K_ff and b_f = -K_fb @ u_c
__global__ void assemble(const int* __restrict__ rows, const int* __restrict__ cols,
                         const float* __restrict__ base_data, const int* __restrict__ elem_idx,
                         const float* __restrict__ sigma,
                         const int* __restrict__ freemap, const int* __restrict__ bmap,
                         const float* __restrict__ u_c,
                         float* __restrict__ K, float* __restrict__ b, int total) {
  int i = blockIdx.x * blockDim.x + threadIdx.x;
  if (i >= total) return;
  int r = rows[i];
  int fr = freemap[r];
  if (fr < 0) return;                       // boundary row: dropped
  int c = cols[i];
  float d = base_data[i] * sigma[elem_idx[i]];
  int fc = freemap[c];
  if (fc >= 0) atomicAdd(&K[(size_t)fr * NPAD + fc], d);
  else         atomicAdd(&b[fr], -d * u_c[bmap[c]]);
}

// ---------------- blocked Cholesky: K = L L^T (lower) ----------------
// Factor 64x64 diagonal block. One workgroup, 64 threads.
__global__ void potf_diag(float* __restrict__ K, int kb) {
  __shared__ float sL[BB * LDP];
  int t = threadIdx.x;
  int base = kb * BB;
  for (int i = 0; i < BB; ++i)
    sL[i * LDP + t] = K[(size_t)(base + i) * NPAD + base + t];
  __syncthreads();
  for (int j = 0; j < BB; ++j) {
    if (t == j) sL[j * LDP + j] = sqrtf(sL[j * LDP + j]);
    __syncthreads();
    float d = sL[j * LDP + j];
    if (t > j) sL[t * LDP + j] /= d;
    __syncthreads();
    if (t > j) {
      float ltj = sL[t * LDP + j];
      for (int k = j + 1; k <= t; ++k)
        sL[t * LDP + k] -= ltj * sL[k * LDP + j];
    }
    __syncthreads();
  }
  for (int i = 0; i < BB; ++i)
    K[(size_t)(base + i) * NPAD + base + t] = sL[i * LDP + t];
}

// Panel TRSM: X * L^T = A for each 64-row block below the diagonal.
__global__ void trsm_panel(float* __restrict__ K, int kb) {
  __shared__ float sL[BB * LDP];
  __shared__ float sA[BB * LDP];
  int t = threadIdx.x;                      // 64 threads; thread = one row of A
  int db = kb * BB;
  int rb = (kb + 1 + blockIdx.x) * BB;
  for (int i = 0; i < BB; ++i) {
    sL[i * LDP + t] = K[(size_t)(db + i) * NPAD + db + t];
    sA[i * LDP + t] = K[(size_t)(rb + i) * NPAD + db + t];
  }
  __syncthreads();
  for (int j = 0; j < BB; ++j) {
    float v = sA[t * LDP + j];
    for (int k = 0; k < j; ++k) v -= sA[t * LDP + k] * sL[j * LDP + k];
    sA[t * LDP + j] = v / sL[j * LDP + j];
  }
  __syncthreads();
  for (int i = 0; i < BB; ++i)
    K[(size_t)(rb + i) * NPAD + db + t] = sA[i * LDP + t];
}

// Trailing SYRK update: C(bi,bj) -= P(bi) * P(bj)^T, lower triangle only.
// 256 threads = 8 waves. Wave w owns sub-tiles (sr,sc) and (sr+2,sc) which share
// B fragments. B panel is negated in LDS and accumulators are seeded with the C
// tile, so D = C + A*(-B) needs no read-modify-write epilogue. All fragments are
// prefetched to registers so the 32 v_wmma_f32_16x16x4_f32 issue back-to-back.
__global__ void syrk_update(float* __restrict__ K, int kb) {
  int bi = kb + 1 + blockIdx.x;             // row block
  int bj = kb + 1 + blockIdx.y;             // col block
  if (bj > bi) return;                      // block-uniform: lower triangle only
  __shared__ float sA[BB * LDP];
  __shared__ float sB[BB * LDP];
  int t = threadIdx.x;
  const int rowA = bi * BB, rowB = bj * BB, colP = kb * BB;
  // stage panels into LDS with b128 ops; negate the B panel
#pragma unroll
  for (int i = 0; i < 4; ++i) {
    int idx = t + i * 256;                  // 0..1023 float4 slots
    int r = idx >> 4, c4 = (idx & 15) * 4;
    float4 va = *(const float4*)&K[(size_t)(rowA + r) * NPAD + colP + c4];
    float4 vb = *(const float4*)&K[(size_t)(rowB + r) * NPAD + colP + c4];
    *(float4*)&sA[r * LDP + c4] = va;
    float4 nb; nb.x = -vb.x; nb.y = -vb.y; nb.z = -vb.z; nb.w = -vb.w;
    *(float4*)&sB[r * LDP + c4] = nb;
  }
  int wave = t >> 5, lane = t & 31;
  int lr = lane & 15, hi = lane >> 4;       // frag: lanes 0-15 hold k,k+1; 16-31 hold k+2,k+3
  int sr = wave >> 2, sc = wave & 3;        // sub-tiles (sr,sc) and (sr+2,sc)
  // seed accumulators with the C tile (loads overlap the LDS fill above)
  float* c1 = K + (size_t)(rowA + sr * 16 + hi * 8) * NPAD + rowB + sc * 16 + lr;
  float* c2 = c1 + (size_t)32 * NPAD;       // sr+2 block of rows
  v8f acc1, acc2;
#pragma unroll
  for (int r = 0; r < 8; ++r) { acc1[r] = c1[(size_t)r * NPAD]; acc2[r] = c2[(size_t)r * NPAD]; }
  __syncthreads();
  // prefetch all fragments for the 64-deep k chain
  v2f bf[16], a1[16], a2[16];
#pragma unroll
  for (int kk = 0; kk < 16; ++kk) {
    int ka = kk * 4 + 2 * hi;
    bf[kk].x = sB[(sc * 16 + lr) * LDP + ka];
    bf[kk].y = sB[(sc * 16 + lr) * LDP + ka + 1];
    a1[kk].x = sA[(sr * 16 + lr) * LDP + ka];
    a1[kk].y = sA[(sr * 16 + lr) * LDP + ka + 1];
    a2[kk].x = sA[((sr + 2) * 16 + lr) * LDP + ka];
    a2[kk].y = sA[((sr + 2) * 16 + lr) * LDP + ka + 1];
  }
  // two independent WMMA chains (shared B) -> pipelined back-to-back issue
#pragma unroll
  for (int kk = 0; kk < 16; ++kk) {
    acc1 = __builtin_amdgcn_wmma_f32_16x16x4_f32(false, a1[kk], false, bf[kk],
                                                 (short)0, acc1, false, false);
    acc2 = __builtin_amdgcn_wmma_f32_16x16x4_f32(false, a2[kk], false, bf[kk],
                                                 (short)0, acc2, false, false);
  }
  // pure-store epilogue (base + immediate row offsets)
#pragma unroll
  for (int r = 0; r < 8; ++r) { c1[(size_t)r * NPAD] = acc1[r]; c2[(size_t)r * NPAD] = acc2[r]; }
}

// ---------------- triangular solves: L y = b, L^T x = y ----------------
__global__ void tri_solve(const float* __restrict__ K, const float* __restrict__ b,
                          float* __restrict__ x) {
  __shared__ float sb[NPAD];                // 25.6 KB of the 320 KB LDS
  int t = threadIdx.x;                      // 1024 threads
  for (int i = t; i < NPAD; i += 1024) sb[i] = b[i];
  __syncthreads();
  for (int j = 0; j < NPAD; ++j) {          // forward: L y = b
    if (t == 0) sb[j] /= K[(size_t)j * NPAD + j];
    __syncthreads();
    float xj = sb[j];
    for (int i = j + 1 + t; i < NPAD; i += 1024)
      sb[i] -= K[(size_t)i * NPAD + j] * xj;
    __syncthreads();
  }
  for (int j = NPAD - 1; j >= 0; --j) {     // backward: L^T x = y  (L^T[i][j] = L[j][i])
    if (t == 0) sb[j] /= K[(size_t)j * NPAD + j];
    __syncthreads();
    float xj = sb[j];
    for (int i = t; i < j; i += 1024)
      sb[i] -= K[(size_t)j * NPAD + i] * xj;
    __syncthreads();
  }
  for (int i = t; i < NPAD; i += 1024) x[i] = sb[i];
}

__global__ void build_u(const int* __restrict__ freemap, const int* __restrict__ bmap,
                        const float* __restrict__ x, const float* __restrict__ u_c,
                        float* __restrict__ u, int nn) {
  int i = blockIdx.x * blockDim.x + threadIdx.x;
  if (i >= nn) return;
  int f = freemap[i];
  u[i] = (f >= 0) ? x[f] : u_c[bmap[i]];
}

// ---------------- energy reduction (deterministic two-stage) ----------------
__global__ void energy_partial(const int* __restrict__ elems, const float* __restrict__ gT,
                               const float* __restrict__ area, const float* __restrict__ sigma,
                               const float* __restrict__ u, float* __restrict__ partial, int E) {
  __shared__ float red[256];
  int e = blockIdx.x * 256 + threadIdx.x;
  float s = 0.f;
  if (e < E) {
    int n0 = elems[3 * e], n1 = elems[3 * e + 1], n2 = elems[3 * e + 2];
    float u0 = u[n0], u1 = u[n1], u2 = u[n2];
    float g0 = gT[6 * e] * u0 + gT[6 * e + 1] * u1 + gT[6 * e + 2] * u2;
    float g1 = gT[6 * e + 3] * u0 + gT[6 * e + 4] * u1 + gT[6 * e + 5] * u2;
    s = sigma[e] * (g0 * g0 + g1 * g1) * area[e];
  }
  red[threadIdx.x] = s;
  __syncthreads();
  for (int o = 128; o > 0; o >>= 1) {
    if (threadIdx.x < o) red[threadIdx.x] += red[threadIdx.x + o];
    __syncthreads();
  }
  if (threadIdx.x == 0) partial[blockIdx.x] = red[0];
}

__global__ void energy_final(const float* __restrict__ partial, float* __restrict__ out, int n) {
  __shared__ float red[64];
  float s = (threadIdx.x < n) ? partial[threadIdx.x] : 0.f;
  red[threadIdx.x] = s;
  __syncthreads();
  for (int o = 32; o > 0; o >>= 1) {
    if (threadIdx.x < o) red[threadIdx.x] += red[threadIdx.x + o];
    __syncthreads();
  }
  if (threadIdx.x == 0) out[0] = red[0];
}

// ---------------- launcher ----------------
extern "C" void kernel_launch(void* const* d_in, const int* in_sizes, int n_in,
                              void* d_out, int out_size, void* d_ws, size_t ws_size,
                              hipStream_t stream) {
  const float* mask      = (const float*)d_in[0];
  const int*   elems     = (const int*)  d_in[1];
  const int*   rows      = (const int*)  d_in[2];
  const int*   cols      = (const int*)  d_in[3];
  const float* base_data = (const float*)d_in[4];
  const int*   elem_idx  = (const int*)  d_in[5];
  const float* gT        = (const float*)d_in[6];
  const float* area      = (const float*)d_in[7];
  const int*   bc_idx    = (const int*)  d_in[8];
  const int*   fc_idx    = (const int*)  d_in[9];
  const float* u_c       = (const float*)d_in[10];

  const int E    = in_sizes[7];             // 12800
  const int nbc  = in_sizes[8];             // 162
  const int nfc  = in_sizes[9];             // 6399
  const int nn   = nfc + nbc;               // 6561
  const int n9   = in_sizes[2];             // 115200 triplet entries

  // workspace layout (16B-aligned chunks)
  char* ws = (char*)d_ws;
  size_t off = 0;
  auto take = [&](size_t bytes) { void* p = ws + off; off += (bytes + 15) & ~(size_t)15; return p; };
  float* Kff     = (float*)take((size_t)NPAD * NPAD * sizeof(float)); // 163.84 MB (fits in L2)
  float* bvec    = (float*)take(NPAD * sizeof(float));
  float* xvec    = (float*)take(NPAD * sizeof(float));
  float* uvec    = (float*)take(nn * sizeof(float));
  float* sigma   = (float*)take(E * sizeof(float));
  int*   freemap = (int*)  take(nn * sizeof(int));
  int*   bmap    = (int*)  take(nn * sizeof(int));
  float* partial = (float*)take(64 * sizeof(float));
  (void)ws_size; (void)n_in; (void)out_size;

  // 1) node maps
  fill_i32<<<(nn + 255) / 256, 256, 0, stream>>>(freemap, -1, nn);
  scatter_map<<<(nfc + 255) / 256, 256, 0, stream>>>(freemap, fc_idx, nfc);
  scatter_map<<<(nbc + 255) / 256, 256, 0, stream>>>(bmap, bc_idx, nbc);

  // 2) per-element conductivity
  sigma_kernel<<<(E + 255) / 256, 256, 0, stream>>>(mask, sigma, E);

  // 3) zero + identity-pad K_ff, zero b
  zero_K<<<((size_t)NPAD * NPAD + 255) / 256, 256, 0, stream>>>(Kff, bvec, nfc);

  // 4) assemble K_ff and b_f
  assemble<<<(n9 + 255) / 256, 256, 0, stream>>>(rows, cols, base_data, elem_idx, sigma,
                                                 freemap, bmap, u_c, Kff, bvec, n9);

  // 5) blocked Cholesky with WMMA f32 trailing updates
  for (int kb = 0; kb < NBLK; ++kb) {
    potf_diag<<<1, BB, 0, stream>>>(Kff, kb);
    int rem = NBLK - kb - 1;
    if (rem > 0) {
      trsm_panel<<<dim3(rem), dim3(BB), 0, stream>>>(Kff, kb);
      syrk_update<<<dim3(rem, rem), dim3(256), 0, stream>>>(Kff, kb);
    }
  }

  // 6) L y = b ; L^T x = y
  tri_solve<<<1, 1024, 0, stream>>>(Kff, bvec, xvec);

  // 7) full solution vector
  build_u<<<(nn + 255) / 256, 256, 0, stream>>>(freemap, bmap, xvec, u_c, uvec, nn);

  // 8) energy (deterministic reduction)
  int nblocks = (E + 255) / 256;            // 50
  energy_partial<<<nblocks, 256, 0, stream>>>(elems, gT, area, sigma, uvec, partial, E);
  energy_final<<<1, 64, 0, stream>>>(partial, (float*)d_out, nblocks);
}